// DyGraphGIN2d_40252433498611
// MI455X (gfx1250) — compile-verified
//
#include <hip/hip_runtime.h>
#include <math.h>

// Problem constants (from reference): B=8, C=64, N=4096, O=64, k=16
#define PN 4096
#define PC 64
#define PB 8
#define PO 64
#define PK 16
#define NROWS (PB * PN)       // 32768 rows through the MLP

typedef float v2f __attribute__((ext_vector_type(2)));
typedef float v8f __attribute__((ext_vector_type(8)));

__device__ __forceinline__ v8f wmma_f32_16x16x4(v2f a, v2f b, v8f c) {
    return __builtin_amdgcn_wmma_f32_16x16x4_f32(
        /*neg_a=*/false, a, /*neg_b=*/false, b,
        /*c_mod=*/(short)0, c, /*reuse_a=*/false, /*reuse_b=*/false);
}

__device__ __forceinline__ float gelu_erf(float x) {
    return 0.5f * x * (1.0f + erff(x * 0.70710678118654752440f));
}

// ---------------------------------------------------------------- kernel 1
// Fused transpose + squared-norm:
//   xf[b][n][c] = x[b][c][n]      (row-major [N][C], 8B-aligned pairs)
//   sq[b*N+n]   = sum_c x[b][c][n]^2
// grid: B * N/64 blocks of 256 threads, 64x64 tile through padded LDS.
__global__ __launch_bounds__(256) void xpose_sqnorm_kernel(const float* __restrict__ x,
                                                           float* __restrict__ xf,
                                                           float* __restrict__ sq) {
    __shared__ float lds[64][65];          // pad -> conflict-free transpose
    const int b  = blockIdx.x >> 6;        // 64 tiles per batch
    const int n0 = (blockIdx.x & 63) * 64;
    const int t  = threadIdx.x;
    const int l  = t & 63;
    const int q  = t >> 6;                 // 0..3
    const float* xb = x + (size_t)b * PC * PN;

#pragma unroll
    for (int i = 0; i < 16; ++i) {         // coalesced loads (n fast)
        int c = i * 4 + q;
        lds[c][l] = xb[(size_t)c * PN + n0 + l];
    }
    __syncthreads();
#pragma unroll
    for (int i = 0; i < 16; ++i) {         // coalesced stores (c fast)
        int n = i * 4 + q;
        xf[((size_t)b * PN + n0 + n) * PC + l] = lds[l][n];
    }
    if (t < 64) {
        float s = 0.f;
#pragma unroll
        for (int c = 0; c < PC; ++c) { float v = lds[c][t]; s += v * v; }
        sq[b * PN + n0 + t] = s;
    }
}

// ---------------------------------------------------------------- kernel 2
// kNN (k=16) via fp32 WMMA Gram tiles + register top-k, then neighbor
// gather-sum:  h[row][c] = (1+eps)*xf[row][c] + sum_nbr xf[nbr][c]
// grid: 256 blocks = B(8) * rowtiles(32 of 128 rows), block = 256 (8 waves)
__global__ __launch_bounds__(256) void knn_aggr_kernel(const float* __restrict__ xf,
                                                       const float* __restrict__ sq,
                                                       const float* __restrict__ epsp,
                                                       float* __restrict__ h) {
    __shared__ float tile[8][16][17];   // per-wave 16x16 Gram tile (padded)
    __shared__ float sqc[16];           // ||x_j||^2 for current column tile
    __shared__ int   nbr[128][16];      // neighbor indices per local row

    const int t    = threadIdx.x;
    const int wave = t >> 5;
    const int lane = t & 31;
    const int m    = lane & 15;         // row/col-in-tile index
    const int half = lane >> 4;
    const int kh   = half * 2;

    const int b  = blockIdx.x >> 5;
    const int rt = blockIdx.x & 31;
    const int rowbase0 = rt * 128;            // block's first row (within batch)
    const int rowbase  = rowbase0 + wave * 16;
    const float* xfb = xf + (size_t)b * PN * PC;
    const float* sqb = sq + b * PN;

    // A fragments: 16 rows x 64 K, resident in 32 VGPRs (b64 loads)
    v2f afrag[16];
    {
        const float* xfr = xfb + (size_t)(rowbase + m) * PC;
#pragma unroll
        for (int kk = 0; kk < 16; ++kk)
            afrag[kk] = *(const v2f*)(xfr + kk * 4 + kh);
    }

    // top-16 state (meaningful on lanes 0..15, one query row per lane)
    float wd[PK];
    int   wi[PK];
#pragma unroll
    for (int i = 0; i < PK; ++i) { wd[i] = 3.4e38f; wi[i] = 0; }
    float worst = 3.4e38f;
    int   wpos  = 0;

    for (int ct = 0; ct < PN / 16; ++ct) {
        const int col0 = ct * 16;
        const float* xfc = xfb + (size_t)(col0 + m) * PC;
        v8f acc = {};
#pragma unroll
        for (int kk = 0; kk < 16; ++kk) {
            v2f bfrag = *(const v2f*)(xfc + kk * 4 + kh);
            acc = wmma_f32_16x16x4(afrag[kk], bfrag, acc);
        }
        if (t < 16) sqc[t] = sqb[col0 + t];
#pragma unroll
        for (int r = 0; r < 8; ++r)
            tile[wave][r + 8 * half][m] = acc[r];
        __syncthreads();

        if (lane < 16) {
#pragma unroll
            for (int j = 0; j < 16; ++j) {
                // d(i,j) ranking key: ||xj||^2 - 2<xi,xj>  (||xi||^2 constant)
                float d = sqc[j] - 2.0f * tile[wave][lane][j];
                if (d < worst) {
                    int cidx = col0 + j;
#pragma unroll
                    for (int i = 0; i < PK; ++i)
                        if (i == wpos) { wd[i] = d; wi[i] = cidx; }
                    worst = wd[0]; wpos = 0;
#pragma unroll
                    for (int i = 1; i < PK; ++i)
                        if (wd[i] > worst) { worst = wd[i]; wpos = i; }
                }
            }
        }
        __syncthreads();   // protect tile/sqc WAR before next iteration
    }

    if (lane < 16) {
#pragma unroll
        for (int j = 0; j < PK; ++j) nbr[wave * 16 + lane][j] = wi[j];
    }
    __syncthreads();

    // gather + sum: thread -> channel c = t&63, rows r = t>>6 step 4
    // xf is [N][C] so lanes sweep a contiguous 256B row -> fully coalesced
    const float eps1 = 1.0f + epsp[0];
    const int c = t & 63;
    for (int r = (t >> 6); r < 128; r += 4) {
        float s = eps1 * xfb[(size_t)(rowbase0 + r) * PC + c];
#pragma unroll
        for (int j = 0; j < PK; ++j)
            s += xfb[(size_t)nbr[r][j] * PC + c];
        h[((size_t)b * PN + rowbase0 + r) * PO + c] = s;   // coalesced (c fast)
    }
}

// ---------------------------------------------------------------- kernel 3
// h1 = h @ w1 + b1 (WMMA) + per-block batchnorm partial sums
// grid: 256 blocks * 128 rows, block 256 (8 waves x 16 rows, 4 col-tiles each)
__global__ __launch_bounds__(256) void gemm1_stats_kernel(const float* __restrict__ h,
                                                          const float* __restrict__ w1,
                                                          const float* __restrict__ b1,
                                                          float* __restrict__ h1,
                                                          float* __restrict__ bpsum,
                                                          float* __restrict__ bpsq) {
    __shared__ float s1[16][64];
    __shared__ float s2[16][64];

    const int t    = threadIdx.x;
    const int wave = t >> 5;
    const int lane = t & 31;
    const int m    = lane & 15;
    const int half = lane >> 4;
    const int kh   = half * 2;
    const int rowbase = blockIdx.x * 128 + wave * 16;

    v2f afrag[16];
    {
        const float* p = h + (size_t)(rowbase + m) * PC + kh;
#pragma unroll
        for (int kk = 0; kk < 16; ++kk)
            afrag[kk] = *(const v2f*)(p + kk * 4);
    }

#pragma unroll
    for (int ct = 0; ct < 4; ++ct) {
        const int col0 = ct * 16;
        v8f acc = {};
#pragma unroll
        for (int kk = 0; kk < 16; ++kk) {
            int k0 = kk * 4 + kh;
            v2f bfrag;
            bfrag.x = w1[(size_t)k0 * PO + col0 + m];
            bfrag.y = w1[(size_t)(k0 + 1) * PO + col0 + m];
            acc = wmma_f32_16x16x4(afrag[kk], bfrag, acc);
        }
        const float bias = b1[col0 + m];
        float ps = 0.f, pq = 0.f;
#pragma unroll
        for (int r = 0; r < 8; ++r) {
            float v = acc[r] + bias;
            h1[(size_t)(rowbase + r + 8 * half) * PO + col0 + m] = v;  // coalesced
            ps += v;
            pq += v * v;
        }
        s1[wave * 2 + half][col0 + m] = ps;
        s2[wave * 2 + half][col0 + m] = pq;
    }
    __syncthreads();

    if (t < 64) {   // deterministic fixed-order reduce of 16 slabs
        float a = 0.f, q = 0.f;
#pragma unroll
        for (int i = 0; i < 16; ++i) { a += s1[i][t]; q += s2[i][t]; }
        bpsum[blockIdx.x * PO + t] = a;
        bpsq [blockIdx.x * PO + t] = q;
    }
}

// ---------------------------------------------------------------- kernel 4
// fold 256 block partials -> fused BN scale/shift per channel
__global__ void stats_reduce_kernel(const float* __restrict__ bpsum,
                                    const float* __restrict__ bpsq,
                                    const float* __restrict__ gamma,
                                    const float* __restrict__ beta,
                                    float* __restrict__ scale,
                                    float* __restrict__ shift) {
    const int c = threadIdx.x;   // 64 threads
    float s = 0.f, q = 0.f;
    for (int i = 0; i < 256; ++i) { s += bpsum[i * PO + c]; q += bpsq[i * PO + c]; }
    const float mean = s * (1.0f / (float)NROWS);
    const float var  = q * (1.0f / (float)NROWS) - mean * mean;
    const float inv  = rsqrtf(var + 1e-5f);
    const float sc   = gamma[c] * inv;
    scale[c] = sc;
    shift[c] = beta[c] - mean * sc;
}

// ---------------------------------------------------------------- kernel 5
// out = GELU(BN(h1)) @ w2 + b2, stored transposed [B, O, N]
__global__ __launch_bounds__(256) void gemm2_out_kernel(const float* __restrict__ h1,
                                                        const float* __restrict__ w2,
                                                        const float* __restrict__ b2,
                                                        const float* __restrict__ scale,
                                                        const float* __restrict__ shift,
                                                        float* __restrict__ out) {
    __shared__ float tileT[8][16][17];  // [wave][o_local][row_local]

    const int t    = threadIdx.x;
    const int wave = t >> 5;
    const int lane = t & 31;
    const int m    = lane & 15;
    const int half = lane >> 4;
    const int kh   = half * 2;
    const int rowbase = blockIdx.x * 128 + wave * 16;
    const int bb    = rowbase >> 12;        // batch
    const int nbase = rowbase & (PN - 1);   // point index base

    // A fragments: b64 load, then BN + erf-GELU applied in registers
    v2f afrag[16];
    {
        const float* p = h1 + (size_t)(rowbase + m) * PO + kh;
#pragma unroll
        for (int kk = 0; kk < 16; ++kk) {
            int k0 = kk * 4 + kh;
            v2f v = *(const v2f*)(p + kk * 4);
            afrag[kk].x = gelu_erf(v.x * scale[k0]     + shift[k0]);
            afrag[kk].y = gelu_erf(v.y * scale[k0 + 1] + shift[k0 + 1]);
        }
    }

#pragma unroll
    for (int ct = 0; ct < 4; ++ct) {
        const int col0 = ct * 16;
        v8f acc = {};
#pragma unroll
        for (int kk = 0; kk < 16; ++kk) {
            int k0 = kk * 4 + kh;
            v2f bfrag;
            bfrag.x = w2[(size_t)k0 * PO + col0 + m];
            bfrag.y = w2[(size_t)(k0 + 1) * PO + col0 + m];
            acc = wmma_f32_16x16x4(afrag[kk], bfrag, acc);
        }
        const float bias = b2[col0 + m];
#pragma unroll
        for (int r = 0; r < 8; ++r)
            tileT[wave][m][r + 8 * half] = acc[r] + bias;
        __syncthreads();
        // transposed store: out[b][o][n], lanes sweep n -> coalesced
#pragma unroll
        for (int rr = 0; rr < 8; ++rr) {
            int ol = half * 8 + rr;
            out[(size_t)bb * PO * PN + (size_t)(col0 + ol) * PN + nbase + m] =
                tileT[wave][ol][m];
        }
        __syncthreads();
    }
}

// ---------------------------------------------------------------- launcher
extern "C" void kernel_launch(void* const* d_in, const int* in_sizes, int n_in,
                              void* d_out, int out_size, void* d_ws, size_t ws_size,
                              hipStream_t stream) {
    const float* x     = (const float*)d_in[0];   // [8,64,4096,1]
    const float* w1    = (const float*)d_in[1];   // [64,64]
    const float* b1    = (const float*)d_in[2];   // [64]
    const float* gamma = (const float*)d_in[3];   // [64]
    const float* beta  = (const float*)d_in[4];   // [64]
    const float* w2    = (const float*)d_in[5];   // [64,64]
    const float* b2    = (const float*)d_in[6];   // [64]
    const float* eps   = (const float*)d_in[7];   // scalar
    (void)in_sizes; (void)n_in; (void)out_size; (void)ws_size;

    float* ws = (float*)d_ws;
    float* sq    = ws;                         // 32768
    float* xfbuf = sq + NROWS;                 // 32768*64  (transposed x)
    float* h     = xfbuf + (size_t)NROWS * PC; // 32768*64
    float* h1    = h + (size_t)NROWS * PO;     // 32768*64
    float* bpsum = h1 + (size_t)NROWS * PO;    // 256*64
    float* bpsq  = bpsum + 256 * PO;           // 256*64
    float* scale = bpsq + 256 * PO;            // 64
    float* shift = scale + PO;                 // 64

    xpose_sqnorm_kernel<<<PB * (PN / 64), 256, 0, stream>>>(x, xfbuf, sq);
    knn_aggr_kernel    <<<PB * (PN / 128), 256, 0, stream>>>(xfbuf, sq, eps, h);
    gemm1_stats_kernel <<<NROWS / 128, 256, 0, stream>>>(h, w1, b1, h1, bpsum, bpsq);
    stats_reduce_kernel<<<1, 64, 0, stream>>>(bpsum, bpsq, gamma, beta, scale, shift);
    gemm2_out_kernel   <<<NROWS / 128, 256, 0, stream>>>(h1, w2, b2, scale, shift,
                                                         (float*)d_out);
}